// RefRectifyNet_14705968022081
// MI455X (gfx1250) — compile-verified
//
#include <hip/hip_runtime.h>
#include <hip/hip_bf16.h>

typedef __attribute__((ext_vector_type(2))) float v2f;
typedef __attribute__((ext_vector_type(8))) float v8f;

#define NB   8
#define CC   256
#define NN   16384   // H*W = 128*128
#define KB   32      // k-chunk staged in LDS
#define LDP  36      // padded LDS row stride (floats): keeps 16B align, avoids bank conflicts

// ---------------------------------------------------------------------------
// Batched cross-Gram GEMM: G[b,i,j] = sum_n x[b,i,n] * y[b,j,n]
// One 256-thread block (8 wave32) per 64x64 tile of G. f32 WMMA 16x16x4.
// ---------------------------------------------------------------------------
__global__ __launch_bounds__(256) void gemm_xy(const float* __restrict__ x,
                                               const float* __restrict__ y,
                                               float* __restrict__ G) {
    __shared__ float As[64][LDP];
    __shared__ float Bs[64][LDP];

    const int b  = blockIdx.z;
    const int i0 = blockIdx.x * 64;
    const int j0 = blockIdx.y * 64;
    const float* xb = x + (size_t)b * CC * NN;
    const float* yb = y + (size_t)b * CC * NN;

    const int t    = threadIdx.x;
    const int wave = t >> 5;       // 0..7
    const int lane = t & 31;
    const int wi   = wave >> 1;    // i-tile 0..3  (16 rows each)
    const int wj   = wave & 1;     // j-half 0..1  (32 cols each -> two 16x16 tiles)

    // global->LDS staging pattern: 8 threads per row, float4 each
    const int lr = t >> 3;          // 0..31
    const int lc = (t & 7) * 4;     // 0..28

    // WMMA fragment addressing (ISA layouts)
    const int fm = lane & 15;            // A: M index / B: N index
    const int kh = (lane >> 4) * 2;      // K pair select: lanes 0-15 -> K 0,1 ; 16-31 -> K 2,3
    const int cn = lane & 15;            // C/D: N
    const int cm = (lane >> 4) * 8;      // C/D: M base (rows v and 8+v)

    v8f acc0 = {};
    v8f acc1 = {};

    for (int k0 = 0; k0 < NN; k0 += KB) {
        __syncthreads();
#pragma unroll
        for (int it = 0; it < 2; ++it) {
            const int r = lr + 32 * it;
            *(float4*)(&As[r][lc]) = *(const float4*)(xb + (size_t)(i0 + r) * NN + k0 + lc);
            *(float4*)(&Bs[r][lc]) = *(const float4*)(yb + (size_t)(j0 + r) * NN + k0 + lc);
        }
        __syncthreads();

#pragma unroll
        for (int kk = 0; kk < KB; kk += 4) {
            v2f a  = *(const v2f*)(&As[wi * 16 + fm][kk + kh]);
            v2f b0 = *(const v2f*)(&Bs[wj * 32 + fm][kk + kh]);
            v2f b1 = *(const v2f*)(&Bs[wj * 32 + 16 + fm][kk + kh]);
            // 8 args: (neg_a, A, neg_b, B, c_mod, C, reuse_a, reuse_b)
            acc0 = __builtin_amdgcn_wmma_f32_16x16x4_f32(false, a, false, b0,
                                                         (short)0, acc0, false, false);
            acc1 = __builtin_amdgcn_wmma_f32_16x16x4_f32(false, a, false, b1,
                                                         (short)0, acc1, false, false);
        }
    }

    // C/D layout: VGPR v, lanes 0-15 -> (M=v, N=lane); lanes 16-31 -> (M=8+v, N=lane-16)
    float* Gb = G + ((size_t)b * CC + i0 + wi * 16 + cm) * CC + j0;
#pragma unroll
    for (int v = 0; v < 8; ++v) {
        Gb[(size_t)v * CC + wj * 32 + cn]      = acc0[v];
        Gb[(size_t)v * CC + wj * 32 + 16 + cn] = acc1[v];
    }
}

// ---------------------------------------------------------------------------
// Row squared-norms of x and y: one block per (b,channel) row.
// ---------------------------------------------------------------------------
__global__ __launch_bounds__(256) void row_norms(const float* __restrict__ x,
                                                 const float* __restrict__ y,
                                                 float* __restrict__ x2,
                                                 float* __restrict__ y2) {
    const int r = blockIdx.x;  // 0..B*CC-1
    const float4* xr = (const float4*)(x + (size_t)r * NN);
    const float4* yr = (const float4*)(y + (size_t)r * NN);
    float sx = 0.f, sy = 0.f;
    for (int k = threadIdx.x; k < NN / 4; k += 256) {
        float4 a = xr[k];
        float4 c = yr[k];
        sx += a.x * a.x + a.y * a.y + a.z * a.z + a.w * a.w;
        sy += c.x * c.x + c.y * c.y + c.z * c.z + c.w * c.w;
    }
    __shared__ float sbx[256], sby[256];
    sbx[threadIdx.x] = sx;
    sby[threadIdx.x] = sy;
    __syncthreads();
    for (int s = 128; s > 0; s >>= 1) {
        if (threadIdx.x < s) {
            sbx[threadIdx.x] += sbx[threadIdx.x + s];
            sby[threadIdx.x] += sby[threadIdx.x + s];
        }
        __syncthreads();
    }
    if (threadIdx.x == 0) { x2[r] = sbx[0]; y2[r] = sby[0]; }
}

// ---------------------------------------------------------------------------
// Per (b,i): dist = sqrt(max(x2+y2-2G,0)); min + first-occurrence argmin over j.
// ---------------------------------------------------------------------------
__global__ __launch_bounds__(256) void min_argmin(const float* __restrict__ G,
                                                  const float* __restrict__ x2,
                                                  const float* __restrict__ y2,
                                                  float* __restrict__ minval,
                                                  int* __restrict__ minidx) {
    const int bi = blockIdx.x;      // b*CC + i
    const int b  = bi >> 8;
    const int j  = threadIdx.x;
    const float g  = G[(size_t)bi * CC + j];
    const float d2 = x2[bi] + y2[b * CC + j] - 2.0f * g;
    const float d  = sqrtf(fmaxf(d2, 0.0f));

    __shared__ float sv[256];
    __shared__ int   si[256];
    sv[j] = d;
    si[j] = j;
    __syncthreads();
    for (int s = 128; s > 0; s >>= 1) {
        if (j < s) {
            const float vo = sv[j + s];
            const int   io = si[j + s];
            if (vo < sv[j] || (vo == sv[j] && io < si[j])) { sv[j] = vo; si[j] = io; }
        }
        __syncthreads();
    }
    if (j == 0) { minval[bi] = sv[0]; minidx[bi] = si[0]; }
}

// ---------------------------------------------------------------------------
// Stable top-128-smallest selection per batch, preserving original channel
// order (rank-by-counting == argsort -> take 128 -> re-sort indices).
// ---------------------------------------------------------------------------
__global__ __launch_bounds__(256) void select_topk(const float* __restrict__ minval,
                                                   const int* __restrict__ minidx,
                                                   int* __restrict__ selidx) {
    const int b = blockIdx.x;
    const int i = threadIdx.x;
    const float* v = minval + b * CC;
    const float vi = v[i];
    int rank = 0;
    for (int j = 0; j < CC; ++j) {
        const float vj = v[j];
        rank += (vj < vi) || (vj == vi && j < i);
    }
    const bool selected = rank < 128;
    __shared__ int selmask[256];
    selmask[i] = selected ? 1 : 0;
    __syncthreads();
    if (selected) {
        int pos = 0;
        for (int j = 0; j < i; ++j) pos += selmask[j];
        selidx[b * 128 + pos] = minidx[b * CC + i];
    }
}

// ---------------------------------------------------------------------------
// Gather selected Ym rows into output: pure streaming copy (HBM-bound).
// ---------------------------------------------------------------------------
__global__ __launch_bounds__(256) void gather_rows(const float* __restrict__ y,
                                                   const int* __restrict__ selidx,
                                                   float* __restrict__ out) {
    const int row = blockIdx.x;     // 0..B*128-1
    const int b   = row >> 7;
    const int p   = row & 127;
    const int src = selidx[b * 128 + p];
    const float4* sp = (const float4*)(y + ((size_t)b * CC + src) * NN);
    float4*       dp = (float4*)(out + ((size_t)b * 128 + p) * NN);
    for (int k = threadIdx.x; k < NN / 4; k += 256) dp[k] = sp[k];
}

// ---------------------------------------------------------------------------
extern "C" void kernel_launch(void* const* d_in, const int* in_sizes, int n_in,
                              void* d_out, int out_size, void* d_ws, size_t ws_size,
                              hipStream_t stream) {
    const float* x = (const float*)d_in[0];
    const float* y = (const float*)d_in[1];
    // d_in[2] = num_matching scalar (always 128 for this problem shape)
    float* out = (float*)d_out;

    char* ws = (char*)d_ws;
    float* G      = (float*)ws;                                  // 8*256*256 f32 = 2 MiB
    float* x2     = (float*)(ws + (size_t)NB * CC * CC * 4);     // 2048 f32
    float* y2     = x2 + NB * CC;                                // 2048 f32
    float* minval = y2 + NB * CC;                                // 2048 f32
    int*   minidx = (int*)(minval + NB * CC);                    // 2048 i32
    int*   selidx = minidx + NB * CC;                            // 1024 i32

    gemm_xy   <<<dim3(CC / 64, CC / 64, NB), 256, 0, stream>>>(x, y, G);
    row_norms <<<NB * CC, 256, 0, stream>>>(x, y, x2, y2);
    min_argmin<<<NB * CC, 256, 0, stream>>>(G, x2, y2, minval, minidx);
    select_topk<<<NB, 256, 0, stream>>>(minval, minidx, selidx);
    gather_rows<<<NB * 128, 256, 0, stream>>>(y, selidx, out);
}